// Gen_71236327572121
// MI455X (gfx1250) — compile-verified
//
#include <hip/hip_runtime.h>

// ---------------------------------------------------------------------------
// Segment softmax over 25M edges (user-grouped with self term, item-grouped).
//
// Roofline: pass1 reads w1,w2,r1,r2 (400MB NT) and scatter-adds exp() into
// sum tables (4MB + 0.8MB -> L2-resident); pass2 re-reads the same 400MB NT,
// recomputes exp (TRANS is free vs. the memory wall), gathers the sums from
// L2 and writes 200MB NT. Total ~1.0GB HBM traffic -> ~43us at 23.3 TB/s.
//
// CDNA5 specifics: wave32 blocks, native global_atomic_add_f32 (non-
// returning -> STOREcnt, resolved at L2), TH_LOAD_NT/TH_STORE_NT hints on
// all streams so they never evict the L2-resident sum tables, v_exp_f32.
// Single-trip kernels (exact grids) so the compiler uses the
// SADDR + 32-bit-voffset scale_offset addressing mode instead of per-lane
// 64-bit address arithmetic.
// ---------------------------------------------------------------------------

typedef float  f4 __attribute__((ext_vector_type(4)));
typedef int    i4 __attribute__((ext_vector_type(4)));

#define BLOCK 256          // 8 wave32 waves per workgroup
#define N_ITEMS_CONST 200000

// ---- probe: first in file so the disasm snippet shows the atomic lowering.
// Never launched; exists only to verify unsafeAtomicAdd -> global_atomic_add_f32.
__global__ void k_atomic_probe(float* __restrict__ p, float v) {
    unsafeAtomicAdd(p + threadIdx.x, v);
}

// ---- pass 1: per-edge exp + scatter-add into L2-resident sum tables -------
__global__ void k_edge_scatter(const float* __restrict__ w1,
                               const float* __restrict__ w2,
                               const int*   __restrict__ r1,
                               const int*   __restrict__ r2,
                               float* __restrict__ sumw1,
                               float* __restrict__ sumw2,
                               int n_edges) {
    const int n4  = n_edges >> 2;
    const int rem = n_edges & 3;
    const int v   = blockIdx.x * BLOCK + threadIdx.x;

    if (v < n4) {
        // 128-bit non-temporal streaming loads (don't pollute L2).
        f4 a  = __builtin_nontemporal_load((const f4*)w1 + v);
        f4 b  = __builtin_nontemporal_load((const f4*)w2 + v);
        i4 iu = __builtin_nontemporal_load((const i4*)r1 + v);
        i4 it = __builtin_nontemporal_load((const i4*)r2 + v);

        f4 ea, eb;
        ea.x = __expf(a.x); ea.y = __expf(a.y);
        ea.z = __expf(a.z); ea.w = __expf(a.w);
        eb.x = __expf(b.x); eb.y = __expf(b.y);
        eb.z = __expf(b.z); eb.w = __expf(b.w);

        // Native FP32 global atomics (no return -> STOREcnt, no wave stall).
        // Targets total 4.8MB -> resolved in L2.
        unsafeAtomicAdd(&sumw1[iu.x], ea.x);
        unsafeAtomicAdd(&sumw1[iu.y], ea.y);
        unsafeAtomicAdd(&sumw1[iu.z], ea.z);
        unsafeAtomicAdd(&sumw1[iu.w], ea.w);
        unsafeAtomicAdd(&sumw2[it.x], eb.x);
        unsafeAtomicAdd(&sumw2[it.y], eb.y);
        unsafeAtomicAdd(&sumw2[it.z], eb.z);
        unsafeAtomicAdd(&sumw2[it.w], eb.w);
    }

    // Scalar tail: first (n_edges & 3) threads of the grid handle one each.
    if (v < rem) {
        const int e = (n4 << 2) + v;
        unsafeAtomicAdd(&sumw1[r1[e]], __expf(w1[e]));
        unsafeAtomicAdd(&sumw2[r2[e]], __expf(w2[e]));
    }
}

// ---- pass 2: recompute exp, gather sums from L2, divide, store ------------
__global__ void k_edge_normalize(const float* __restrict__ w1,
                                 const float* __restrict__ w2,
                                 const int*   __restrict__ r1,
                                 const int*   __restrict__ r2,
                                 const float* __restrict__ sumw1,
                                 const float* __restrict__ sumw2,
                                 float* __restrict__ lfw1,
                                 float* __restrict__ lfw2,
                                 int n_edges) {
    const int n4  = n_edges >> 2;
    const int rem = n_edges & 3;
    const int v   = blockIdx.x * BLOCK + threadIdx.x;

    if (v < n4) {
        f4 a  = __builtin_nontemporal_load((const f4*)w1 + v);
        f4 b  = __builtin_nontemporal_load((const f4*)w2 + v);
        i4 iu = __builtin_nontemporal_load((const i4*)r1 + v);
        i4 it = __builtin_nontemporal_load((const i4*)r2 + v);

        f4 ea, eb;
        ea.x = __expf(a.x); ea.y = __expf(a.y);
        ea.z = __expf(a.z); ea.w = __expf(a.w);
        eb.x = __expf(b.x); eb.y = __expf(b.y);
        eb.z = __expf(b.z); eb.w = __expf(b.w);

        // Gathers hit the L2-resident sum tables (regular temporal hint).
        f4 sa, sb;
        sa.x = sumw1[iu.x]; sa.y = sumw1[iu.y];
        sa.z = sumw1[iu.z]; sa.w = sumw1[iu.w];
        sb.x = sumw2[it.x]; sb.y = sumw2[it.y];
        sb.z = sumw2[it.z]; sb.w = sumw2[it.w];

        f4 oa, ob;
        oa.x = ea.x / sa.x; oa.y = ea.y / sa.y;
        oa.z = ea.z / sa.z; oa.w = ea.w / sa.w;
        ob.x = eb.x / sb.x; ob.y = eb.y / sb.y;
        ob.z = eb.z / sb.z; ob.w = eb.w / sb.w;

        __builtin_nontemporal_store(oa, (f4*)lfw1 + v);
        __builtin_nontemporal_store(ob, (f4*)lfw2 + v);
    }

    if (v < rem) {
        const int e = (n4 << 2) + v;
        lfw1[e] = __expf(w1[e]) / sumw1[r1[e]];
        lfw2[e] = __expf(w2[e]) / sumw2[r2[e]];
    }
}

// ---- init: sumw1[u] = exp(w0[u]) (stash expw0 too), sumw2[i] = 0 ----------
__global__ void k_init(const float* __restrict__ w0,
                       float* __restrict__ sumw1,
                       float* __restrict__ expw0,
                       float* __restrict__ sumw2,
                       int n_users, int n_items) {
    int i = blockIdx.x * BLOCK + threadIdx.x;
    if (i < n_users) {
        float e = __expf(w0[i]);
        expw0[i] = e;
        sumw1[i] = e;
    }
    if (i < n_items) {
        sumw2[i] = 0.0f;
    }
}

// ---- lfw0[u] = expw0[u] / sumw1[u] ----------------------------------------
__global__ void k_user(const float* __restrict__ expw0,
                       const float* __restrict__ sumw1,
                       float* __restrict__ lfw0,
                       int n_users) {
    int i = blockIdx.x * BLOCK + threadIdx.x;
    if (i < n_users) {
        lfw0[i] = expw0[i] / sumw1[i];
    }
}

extern "C" void kernel_launch(void* const* d_in, const int* in_sizes, int n_in,
                              void* d_out, int out_size, void* d_ws, size_t ws_size,
                              hipStream_t stream) {
    const float* w0 = (const float*)d_in[0];
    const float* w1 = (const float*)d_in[1];
    const float* w2 = (const float*)d_in[2];
    const int*   r1 = (const int*)d_in[3];
    const int*   r2 = (const int*)d_in[4];

    const int n_users = in_sizes[0];
    const int n_edges = in_sizes[1];
    const int n_items = N_ITEMS_CONST;   // fixed by the reference problem

    // Output layout: [lfw0 (n_users)] [lfw1 (n_edges)] [lfw2 (n_edges)]
    float* lfw0 = (float*)d_out;
    float* lfw1 = lfw0 + n_users;
    float* lfw2 = lfw1 + n_edges;

    // Workspace layout: sumw1[n_users] | sumw2[n_items] | expw0[n_users]
    float* sumw1 = (float*)d_ws;
    float* sumw2 = sumw1 + n_users;
    float* expw0 = sumw2 + n_items;

    const int init_grid = (n_users + BLOCK - 1) / BLOCK;   // n_users >= n_items
    const int n4        = n_edges >> 2;
    int edge_grid       = (n4 + BLOCK - 1) / BLOCK;
    if (edge_grid < 1) edge_grid = 1;

    k_init<<<init_grid, BLOCK, 0, stream>>>(w0, sumw1, expw0, sumw2,
                                            n_users, n_items);

    k_edge_scatter<<<edge_grid, BLOCK, 0, stream>>>(w1, w2, r1, r2,
                                                    sumw1, sumw2, n_edges);

    k_edge_normalize<<<edge_grid, BLOCK, 0, stream>>>(w1, w2, r1, r2,
                                                      sumw1, sumw2,
                                                      lfw1, lfw2, n_edges);

    k_user<<<init_grid, BLOCK, 0, stream>>>(expw0, sumw1, lfw0, n_users);
}